// DRP_LAYER_89240830476984
// MI455X (gfx1250) — compile-verified
//
#include <hip/hip_runtime.h>

#define NN   1024
#define HXR  1022
#define HXC  1023
#define HYR  1023
#define HYC  1022

typedef __attribute__((ext_vector_type(2))) float v2f;
typedef __attribute__((ext_vector_type(8))) float v8f;
typedef __attribute__((ext_vector_type(4))) unsigned int v4u;
typedef __attribute__((ext_vector_type(8))) int v8i_t;
typedef __attribute__((ext_vector_type(4))) int v4i_t;

#if defined(__has_builtin)
# if __has_builtin(__builtin_amdgcn_wmma_f32_16x16x4_f32)
#  define USE_WMMA 1
# endif
# if __has_builtin(__builtin_amdgcn_global_load_async_to_lds_b32) && \
     __has_builtin(__builtin_amdgcn_s_wait_asynccnt)
#  define USE_ASYNC_LDS 1
# endif
# if __has_builtin(__builtin_amdgcn_tensor_load_to_lds) && \
     __has_builtin(__builtin_amdgcn_s_wait_tensorcnt)
#  define USE_TDM 1
# endif
#endif
#ifndef USE_WMMA
# define USE_WMMA 0
#endif
#ifndef USE_ASYNC_LDS
# define USE_ASYNC_LDS 0
#endif
#ifndef USE_TDM
# define USE_TDM 0
#endif

#if USE_ASYNC_LDS
// param0 = global source ptr (addrspace(1) int*, prints as "__device__ int *"),
// param1 = LDS dest ptr, then imm offset and imm cpol. Clang rewrites pointer
// param address spaces to match the arguments.
# define STAGE_F32(ldsp, gp)                                                   \
    __builtin_amdgcn_global_load_async_to_lds_b32((int*)(gp), (int*)(ldsp), 0, 0)
# define STAGE_FENCE() __builtin_amdgcn_s_wait_asynccnt(0)
#else
# define STAGE_F32(ldsp, gp) (*(ldsp) = *(gp))
# define STAGE_FENCE() ((void)0)
#endif

__device__ __forceinline__ float bsel(int d, float q0, float q1, float q2, float q3) {
  float r = 0.0f;
  r = (d == 0) ? q0 : r;
  r = (d == 1) ? q1 : r;
  r = (d == 2) ? q2 : r;
  r = (d == 3) ? q3 : r;
  return r;
}

#if USE_WMMA
__device__ __forceinline__ v8f wmma_k4(v2f a, v2f b, v8f c) {
  // V_WMMA_F32_16X16X4_F32 : D = A(16x4) x B(4x16) + C(16x16), fp32
  return __builtin_amdgcn_wmma_f32_16x16x4_f32(false, a, false, b, (short)0, c,
                                               false, false);
}
#endif

// ---------------- scalar (boundary) evaluators -------------------------------

__device__ float amper_point(const float* __restrict__ Eb,
                             const float* __restrict__ Hxb,
                             const float* __restrict__ Hyb,
                             int r, int c,
                             float q0, float q1, float q2, float q3) {
  const float i24 = 1.0f / 24.0f;
  const float kf0 = -23.f*i24, kf1 = 21.f*i24, kf2 = 3.f*i24,  kf3 = -1.f*i24;
  const float kb0 = 1.f*i24,   kb1 = -3.f*i24, kb2 = -21.f*i24, kb3 = 23.f*i24;
  const float up0 = 1.f*i24,   up1 = -27.f*i24, up2 = 27.f*i24, up3 = -1.f*i24;
  const float dn0 = -1.f*i24,  dn1 = 27.f*i24, dn2 = -27.f*i24, dn3 = 1.f*i24;

  float s1 = 0.f, s2 = 0.f;
  const bool rin = (r >= 2 && r <= NN - 3);
  const bool cin = (c >= 2 && c <= NN - 3);
  if (rin && cin) {
    const float* p = Hyb + (r - 2) * HYC + (c - 2);
    s1 += q0 * (p[1] - p[0]); p += HYC;
    s1 += q1 * (p[1] - p[0]); p += HYC;
    s1 += q2 * (p[1] - p[0]); p += HYC;
    s1 += q3 * (p[1] - p[0]);
    const float* a = Hxb + (r - 2) * HXC + (c - 2);
    const float* b = a + HXC;
    s2 += q0 * (b[0] - a[0]) + q1 * (b[1] - a[1]) + q2 * (b[2] - a[2]) + q3 * (b[3] - a[3]);
  }
  if (c == 1) {
    if (r <= HYR - 1) {
      const float* p = Hyb + r * HYC;
      s1 += kf0 * p[0] + kf1 * p[1] + kf2 * p[2] + kf3 * p[3];
    }
    if (rin) {
      const float* p = Hyb + (r - 2) * HYC;
      s1 += up0 * p[0] + up1 * p[HYC] + up2 * p[2 * HYC] + up3 * p[3 * HYC];
    }
  }
  if (c == HYC) {  // c == 1022
    if (r >= 1) {
      const float* p = Hyb + (r - 1) * HYC + (HYC - 4);
      s1 += kb0 * p[0] + kb1 * p[1] + kb2 * p[2] + kb3 * p[3];
    }
    if (rin) {
      const float* p = Hyb + (r - 2) * HYC + (HYC - 1);
      s1 += dn0 * p[0] + dn1 * p[HYC] + dn2 * p[2 * HYC] + dn3 * p[3 * HYC];
    }
  }
  if (r == 1) {
    if (c >= 1) {
      const float* p = Hxb + (c - 1);
      s2 += kf0 * p[0] + kf1 * p[HXC] + kf2 * p[2 * HXC] + kf3 * p[3 * HXC];
    }
    if (cin) {
      const float* p = Hxb + (c - 2);
      s2 += up0 * p[0] + up1 * p[1] + up2 * p[2] + up3 * p[3];
    }
  }
  if (r == HXR) {  // r == 1022
    if (c <= HXC - 1) {
      const float* p = Hxb + (HXR - 4) * HXC + c;
      s2 += kb0 * p[0] + kb1 * p[HXC] + kb2 * p[2 * HXC] + kb3 * p[3 * HXC];
    }
    if (cin) {
      const float* p = Hxb + (HXR - 1) * HXC + (c - 2);
      s2 += dn0 * p[0] + dn1 * p[1] + dn2 * p[2] + dn3 * p[3];
    }
  }
  return Eb[r * NN + c] + 0.5f * (s1 - s2);
}

__device__ float far_hx_point(const float* __restrict__ Eb,
                              const float* __restrict__ Hxb,
                              int i, int j,
                              float q0, float q1, float q2, float q3) {
  float s3 = 0.f;
  if (j >= 1 && j <= NN - 3) {
    const float* a = Eb + i * NN + (j - 1);
    const float* b = a + NN;
    s3 = q0 * (b[0] - a[0]) + q1 * (b[1] - a[1]) + q2 * (b[2] - a[2]) + q3 * (b[3] - a[3]);
  }
  if (i == 0)        { const float* p = Eb + 1 * NN + j;        s3 += p[1] - p[0]; }
  if (i == HXR - 1)  { const float* p = Eb + (NN - 2) * NN + j; s3 += p[1] - p[0]; }
  return Hxb[i * HXC + j] - 0.5f * s3;
}

__device__ float far_hy_point(const float* __restrict__ Eb,
                              const float* __restrict__ Hyb,
                              int i, int j,
                              float q0, float q1, float q2, float q3) {
  float s4 = 0.f;
  if (i >= 1 && i <= NN - 3) {
    const float* p = Eb + (i - 1) * NN + j;
    s4 += q0 * (p[1] - p[0]); p += NN;
    s4 += q1 * (p[1] - p[0]); p += NN;
    s4 += q2 * (p[1] - p[0]); p += NN;
    s4 += q3 * (p[1] - p[0]);
  }
  if (j == 0)        { const float* p = Eb + i * NN + 1;        s4 += p[NN] - p[0]; }
  if (j == HYC - 1)  { const float* p = Eb + i * NN + (NN - 2); s4 += p[NN] - p[0]; }
  return Hyb[i * HYC + j] + 0.5f * s4;
}

// ---------------- fused passes ----------------------------------------------
// Block = 256 threads = 8 waves covering a 32x64 region; each wave owns one
// 16x16 output tile in the WMMA C/D register layout. Interior tiles use
// banded-matrix WMMA (5 chained V_WMMA_F32_16X16X4_F32 per conv direction),
// data staged to LDS via TDM (TENSOR_LOAD_TO_LDS, interior blocks of faraday)
// or GLOBAL_LOAD_ASYNC_TO_LDS. Boundary tiles take a wave-uniform scalar path.

__global__ __launch_bounds__(256) void fdtd_amper(
    const float* __restrict__ E, const float* __restrict__ Hx,
    const float* __restrict__ Hy,
    const float* __restrict__ pbeta, const float* __restrict__ pdelta,
    const float* __restrict__ pgamma, float* __restrict__ Eo) {
  const int b  = blockIdx.z;
  const int rb = blockIdx.y * 32;
  const int cb = blockIdx.x * 64;
  const float* Eb  = E  + (size_t)b * NN * NN;
  const float* Hxb = Hx + (size_t)b * HXR * HXC;
  const float* Hyb = Hy + (size_t)b * HYR * HYC;
  float* Eob = Eo + (size_t)b * NN * NN;

  const float beta = pbeta[0], delta = pdelta[0], gamma = pgamma[0];
  const float i24 = 1.0f / 24.0f;
  const float q0 = beta * i24;
  const float q1 = 1.0f + (delta - 2.0f * beta) * i24;
  const float q2 = (beta + gamma) * i24;
  const float q3 = -(delta + gamma) * i24;

  // Hx halo tile: rows rb-2..rb+30, cols cb-2..cb+65 (clamped).
  __shared__ float sHx[33][68];
  // Hy halo tile: rows rb-2..rb+33, cols cb-2..cb+65 (clamped).
  __shared__ float sHy[36][68];

  for (int idx = threadIdx.x; idx < 33 * 68; idx += 256) {
    int rr = idx / 68, cc = idx - rr * 68;
    int gr = rb - 2 + rr; gr = gr < 0 ? 0 : (gr > HXR - 1 ? HXR - 1 : gr);
    int gc = cb - 2 + cc; gc = gc < 0 ? 0 : (gc > HXC - 1 ? HXC - 1 : gc);
    STAGE_F32(&sHx[rr][cc], Hxb + gr * HXC + gc);
  }
  for (int idx = threadIdx.x; idx < 36 * 68; idx += 256) {
    int rr = idx / 68, cc = idx - rr * 68;
    int gr = rb - 2 + rr; gr = gr < 0 ? 0 : (gr > HYR - 1 ? HYR - 1 : gr);
    int gc = cb - 2 + cc; gc = gc < 0 ? 0 : (gc > HYC - 1 ? HYC - 1 : gc);
    STAGE_F32(&sHy[rr][cc], Hyb + gr * HYC + gc);
  }
  STAGE_FENCE();
  __syncthreads();

  const int wave = threadIdx.x >> 5;
  const int lane = threadIdx.x & 31;
  const int half = lane >> 4;
  const int nn   = lane & 15;
  const int lr = (wave >> 2) << 4;
  const int lc = (wave & 3) << 4;
  const int r0 = rb + lr;
  const int c0 = cb + lc;

#if USE_WMMA
  const bool wok = (r0 >= 16 && r0 <= 992 && c0 >= 16 && c0 <= 992);
  if (wok) {
    // banded coefficient operand (same formula serves A- and B-side bands)
    v2f bwP[5], bwN[5];
#pragma unroll
    for (int kk = 0; kk < 5; ++kk) {
      const int k0 = 4 * kk + 2 * half;
      v2f t;
      t.x = bsel(k0 - nn,     q0, q1, q2, q3);
      t.y = bsel(k0 + 1 - nn, q0, q1, q2, q3);
      bwP[kk] = t;
      bwN[kk] = -t;
    }
    v8f acc = {0.f, 0.f, 0.f, 0.f, 0.f, 0.f, 0.f, 0.f};
    // s1: 4-tap along rows of (Hy col-diff). A = band, B = data from LDS
#pragma unroll
    for (int kk = 0; kk < 5; ++kk) {
      const int k0 = 4 * kk + 2 * half;
      v2f bb;
      bb.x = sHy[lr + k0][lc + nn + 1]     - sHy[lr + k0][lc + nn];
      bb.y = sHy[lr + k0 + 1][lc + nn + 1] - sHy[lr + k0 + 1][lc + nn];
      acc = wmma_k4(bwP[kk], bb, acc);
    }
    // s2 (subtracted via negated band): 4-tap along cols of (Hx row-diff)
    const int rA = lr + nn;
#pragma unroll
    for (int kk = 0; kk < 5; ++kk) {
      const int k0 = 4 * kk + 2 * half;
      v2f aa;
      aa.x = sHx[rA + 1][lc + k0]     - sHx[rA][lc + k0];
      aa.y = sHx[rA + 1][lc + k0 + 1] - sHx[rA][lc + k0 + 1];
      acc = wmma_k4(aa, bwN[kk], acc);
    }
#pragma unroll
    for (int v = 0; v < 8; ++v) {
      const int r = r0 + (half << 3) + v;
      const int c = c0 + nn;
      Eob[r * NN + c] = Eb[r * NN + c] + 0.5f * acc[v];
    }
  } else
#endif
  {
#pragma unroll
    for (int v = 0; v < 8; ++v) {
      const int r = r0 + (half << 3) + v;
      const int c = c0 + nn;
      Eob[r * NN + c] = amper_point(Eb, Hxb, Hyb, r, c, q0, q1, q2, q3);
    }
  }
}

// Fused faraday: both H updates read only E; one LDS tile serves both.
__global__ __launch_bounds__(256) void fdtd_faraday(
    const float* __restrict__ E, const float* __restrict__ Hx,
    const float* __restrict__ Hy,
    const float* __restrict__ pbeta, const float* __restrict__ pdelta,
    const float* __restrict__ pgamma,
    float* __restrict__ Hxo, float* __restrict__ Hyo) {
  const int b  = blockIdx.z;
  const int rb = blockIdx.y * 32;
  const int cb = blockIdx.x * 64;
  const float* Eb  = E  + (size_t)b * NN * NN;
  const float* Hxb = Hx + (size_t)b * HXR * HXC;
  const float* Hyb = Hy + (size_t)b * HYR * HYC;
  float* Hxob = Hxo + (size_t)b * HXR * HXC;
  float* Hyob = Hyo + (size_t)b * HYR * HYC;

  const float beta = pbeta[0], delta = pdelta[0], gamma = pgamma[0];
  const float i24 = 1.0f / 24.0f;
  const float q0 = beta * i24;
  const float q1 = 1.0f + (delta - 2.0f * beta) * i24;
  const float q2 = (beta + gamma) * i24;
  const float q3 = -(delta + gamma) * i24;

  // E tile: rows rb-1..rb+34, cols cb-1..cb+66 (clamped / TDM when interior).
  __shared__ float sE[36][68];

#if USE_TDM
  // Fully interior block: one TDM descriptor moves the whole 36x68 f32 tile.
  const bool tdm_ok = (rb >= 1) && (rb + 35 < NN) && (cb >= 1) && (cb + 67 < NN);
  if (tdm_ok) {
    if (threadIdx.x < 32) {  // wave 0 issues the tensor DMA (per-wave issue)
      const unsigned long long ga =
          (unsigned long long)(uintptr_t)(Eb + (size_t)(rb - 1) * NN + (cb - 1));
      v4u g0;
      g0.x = 1u;                                   // count=1 (valid user D#)
      g0.y = (unsigned)(uintptr_t)(&sE[0][0]);     // lds_addr (byte offset)
      g0.z = (unsigned)ga;                         // global_addr[31:0]
      g0.w = (unsigned)((ga >> 32) & 0x01FFFFFFu)  // global_addr[56:32]
             | (2u << 30);                         // type = 2 ("image")
      v8i_t g1;
      g1[0] = 0x00020000;          // data_size=2 (4 bytes); no multicast/pad
      g1[1] = (int)(68u << 16);    // tensor_dim0 = 68 (lo 16 in [31:16])
      g1[2] = (int)(36u << 16);    // tensor_dim0 hi = 0 ; tensor_dim1 = 36 (lo)
      g1[3] = (int)(68u << 16);    // tensor_dim1 hi = 0 ; tile_dim0 = 68
      g1[4] = 36;                  // tile_dim1 = 36 ; tile_dim2 = 0
      g1[5] = NN;                  // tensor_dim0_stride[31:0] = 1024
      g1[6] = 0;                   // stride0 hi ; tensor_dim1_stride lo
      g1[7] = 0;                   // tensor_dim1_stride hi
      v4i_t g2 = {0, 0, 0, 0};
      v4i_t g3 = {0, 0, 0, 0};
      v8i_t g4 = {0, 0, 0, 0, 0, 0, 0, 0};  // extra group (6-arg builtin form)
      __builtin_amdgcn_tensor_load_to_lds(g0, g1, g2, g3, g4, 0);
      __builtin_amdgcn_s_wait_tensorcnt(0);
    }
  } else
#endif
  {
    for (int idx = threadIdx.x; idx < 36 * 68; idx += 256) {
      int rr = idx / 68, cc = idx - rr * 68;
      int gr = rb - 1 + rr; gr = gr < 0 ? 0 : (gr > NN - 1 ? NN - 1 : gr);
      int gc = cb - 1 + cc; gc = gc < 0 ? 0 : (gc > NN - 1 ? NN - 1 : gc);
      STAGE_F32(&sE[rr][cc], Eb + gr * NN + gc);
    }
    STAGE_FENCE();
  }
  __syncthreads();

  const int wave = threadIdx.x >> 5;
  const int lane = threadIdx.x & 31;
  const int half = lane >> 4;
  const int nn   = lane & 15;
  const int lr = (wave >> 2) << 4;
  const int lc = (wave & 3) << 4;
  const int r0 = rb + lr;
  const int c0 = cb + lc;

#if USE_WMMA
  const bool wok = (r0 >= 16 && r0 <= 992 && c0 >= 16 && c0 <= 992);
  if (wok) {
    v2f bw[5];
#pragma unroll
    for (int kk = 0; kk < 5; ++kk) {
      const int k0 = 4 * kk + 2 * half;
      v2f t;
      t.x = bsel(k0 - nn,     q0, q1, q2, q3);
      t.y = bsel(k0 + 1 - nn, q0, q1, q2, q3);
      bw[kk] = t;
    }
    // ---- Hx: 4-tap along cols of (E row-diff); A = data, B = band
    v8f acc = {0.f, 0.f, 0.f, 0.f, 0.f, 0.f, 0.f, 0.f};
    const int rA = lr + nn;
#pragma unroll
    for (int kk = 0; kk < 5; ++kk) {
      const int k0 = 4 * kk + 2 * half;
      v2f aa;
      aa.x = sE[rA + 2][lc + k0]     - sE[rA + 1][lc + k0];
      aa.y = sE[rA + 2][lc + k0 + 1] - sE[rA + 1][lc + k0 + 1];
      acc = wmma_k4(aa, bw[kk], acc);
    }
    // ---- Hy: 4-tap along rows of (E col-diff); A = band, B = data
    v8f acc2 = {0.f, 0.f, 0.f, 0.f, 0.f, 0.f, 0.f, 0.f};
#pragma unroll
    for (int kk = 0; kk < 5; ++kk) {
      const int k0 = 4 * kk + 2 * half;
      v2f bb;
      bb.x = sE[lr + k0][lc + nn + 2]     - sE[lr + k0][lc + nn + 1];
      bb.y = sE[lr + k0 + 1][lc + nn + 2] - sE[lr + k0 + 1][lc + nn + 1];
      acc2 = wmma_k4(bw[kk], bb, acc2);
    }
#pragma unroll
    for (int v = 0; v < 8; ++v) {
      const int r = r0 + (half << 3) + v;
      const int c = c0 + nn;
      Hxob[r * HXC + c] = Hxb[r * HXC + c] - 0.5f * acc[v];
      Hyob[r * HYC + c] = Hyb[r * HYC + c] + 0.5f * acc2[v];
    }
  } else
#endif
  {
#pragma unroll
    for (int v = 0; v < 8; ++v) {
      const int r = r0 + (half << 3) + v;
      const int c = c0 + nn;
      if (r < HXR && c < HXC)
        Hxob[r * HXC + c] = far_hx_point(Eb, Hxb, r, c, q0, q1, q2, q3);
      if (r < HYR && c < HYC)
        Hyob[r * HYC + c] = far_hy_point(Eb, Hyb, r, c, q0, q1, q2, q3);
    }
  }
}

// ---------------- launch -----------------------------------------------------

extern "C" void kernel_launch(void* const* d_in, const int* in_sizes, int n_in,
                              void* d_out, int out_size, void* d_ws, size_t ws_size,
                              hipStream_t stream) {
  (void)in_sizes; (void)n_in; (void)out_size; (void)d_ws; (void)ws_size;
  const float* E1  = (const float*)d_in[0];
  const float* Hx1 = (const float*)d_in[1];
  const float* Hy1 = (const float*)d_in[2];
  const float* pb  = (const float*)d_in[9];
  const float* pd  = (const float*)d_in[10];
  const float* pg  = (const float*)d_in[11];

  float* out = (float*)d_out;
  const size_t NE = (size_t)4 * NN * NN;
  const size_t NH = (size_t)4 * HXR * HXC;  // == 4*HYR*HYC
  float* E2  = out;
  float* Hx2 = E2 + NE;
  float* Hy2 = Hx2 + NH;
  float* E3  = Hy2 + NH;
  float* Hx3 = E3 + NE;
  float* Hy3 = Hx3 + NH;
  float* E4  = Hy3 + NH;
  float* Hx4 = E4 + NE;
  float* Hy4 = Hx4 + NH;

  dim3 blk(256, 1, 1);
  dim3 grd(16, 32, 4);  // 64-col x 32-row blocks x batch

  fdtd_amper<<<grd, blk, 0, stream>>>(E1, Hx1, Hy1, pb, pd, pg, E2);
  fdtd_faraday<<<grd, blk, 0, stream>>>(E2, Hx1, Hy1, pb, pd, pg, Hx2, Hy2);

  fdtd_amper<<<grd, blk, 0, stream>>>(E2, Hx2, Hy2, pb, pd, pg, E3);
  fdtd_faraday<<<grd, blk, 0, stream>>>(E3, Hx2, Hy2, pb, pd, pg, Hx3, Hy3);

  fdtd_amper<<<grd, blk, 0, stream>>>(E3, Hx3, Hy3, pb, pd, pg, E4);
  fdtd_faraday<<<grd, blk, 0, stream>>>(E4, Hx3, Hy3, pb, pd, pg, Hx4, Hy4);
}